// termination_predictor_35734127903068
// MI455X (gfx1250) — compile-verified
//
#include <hip/hip_runtime.h>
#include <hip/hip_bf16.h>

#define N_NODES  200000
#define N_EDGES  3200000
#define N_FEAT   128
#define N_GRAPHS 1024

typedef float v2f __attribute__((ext_vector_type(2)));
typedef float v8f __attribute__((ext_vector_type(8)));

// ---------------------------------------------------------------------------
// Degree / normalization
// ---------------------------------------------------------------------------
__global__ void __launch_bounds__(256)
deg_kernel(const int* __restrict__ ei, float* __restrict__ deg) {
    int e = blockIdx.x * 256 + threadIdx.x;
    if (e >= N_EDGES) return;
    atomicAdd(&deg[ei[N_EDGES + e]], 1.0f);
}

__global__ void __launch_bounds__(256)
dinv_kernel(float* __restrict__ deg) {
    int n = blockIdx.x * 256 + threadIdx.x;
    if (n >= N_NODES) return;
    deg[n] = rsqrtf(deg[n] + 1.0f);   // in place: deg -> deg^{-1/2}
}

// ---------------------------------------------------------------------------
// WMMA GEMM over node features:  Y[n, 0..63] = X[n, 0..K-1] @ Wpad[K, 64]
// W (wrows x wcols, row-major) is staged in LDS pre-swizzled as K-pairs:
//   Wl2[k/2][c] = { W[k][c], W[k+1][c] }  (zero padded)
// so each 4x16 B fragment is ONE aligned ds_load_b64 into an even VGPR pair.
// One wave computes one 16-row M tile x 64 columns via V_WMMA_F32_16X16X4_F32.
// Pad columns are exactly 0 -> reused as K padding for the next GEMM.
// ---------------------------------------------------------------------------
__global__ void __launch_bounds__(256)
wmma_gemm_node(const float* __restrict__ X, int ldx, int K,
               const float* __restrict__ W, int wrows, int wcols,
               float* __restrict__ Y, int n_nodes)
{
    __shared__ v2f Wl2[64 * 64];             // max K/2=64 pairs x 64 cols = 32 KB
    const int tid = threadIdx.x;
    const int pairs = (K >> 1) * 64;
    for (int i = tid; i < pairs; i += 256) {
        int p = i >> 6, c = i & 63;
        int k = p << 1;
        v2f w;
        w.x = (k     < wrows && c < wcols) ? W[(size_t)k * wcols + c]       : 0.0f;
        w.y = (k + 1 < wrows && c < wcols) ? W[(size_t)(k + 1) * wcols + c] : 0.0f;
        Wl2[i] = w;
    }
    __syncthreads();

    const int wave  = tid >> 5;
    const int lane  = tid & 31;
    const int m     = lane & 15;             // M index (A) / N index (B,C,D)
    const int half  = lane >> 4;             // selects K pair / upper M rows
    const int mtile = blockIdx.x * 8 + wave;
    if (mtile * 16 >= n_nodes) return;

    const float* xrow = X + (size_t)(mtile * 16 + m) * (size_t)ldx;

    v8f acc0 = {}, acc1 = {}, acc2 = {}, acc3 = {};
    for (int k0 = 0; k0 < K; k0 += 4) {
        const int kb = k0 + 2 * half;        // even -> 8B aligned
        v2f a = *(const v2f*)(xrow + kb);    // A[m][kb], A[m][kb+1]

        const v2f* wp = &Wl2[(size_t)(kb >> 1) * 64];
        v2f b0 = wp[m];
        v2f b1 = wp[16 + m];
        v2f b2 = wp[32 + m];
        v2f b3 = wp[48 + m];

        acc0 = __builtin_amdgcn_wmma_f32_16x16x4_f32(false, a, false, b0, (short)0, acc0, false, false);
        acc1 = __builtin_amdgcn_wmma_f32_16x16x4_f32(false, a, false, b1, (short)0, acc1, false, false);
        acc2 = __builtin_amdgcn_wmma_f32_16x16x4_f32(false, a, false, b2, (short)0, acc2, false, false);
        acc3 = __builtin_amdgcn_wmma_f32_16x16x4_f32(false, a, false, b3, (short)0, acc3, false, false);
    }

    float* yb = Y + (size_t)(mtile * 16) * 64;
    #pragma unroll
    for (int i = 0; i < 8; ++i) {
        const size_t r = (size_t)(i + 8 * half) * 64;
        yb[r + m]      = acc0[i];
        yb[r + 16 + m] = acc1[i];
        yb[r + 32 + m] = acc2[i];
        yb[r + 48 + m] = acc3[i];
    }
}

// ---------------------------------------------------------------------------
// Self-loop + bias INIT:  Hout[n,c] = Hin[n,c]*dinv[n]^2 + b[c]
// Runs BEFORE the edge scatter (which then accumulates on top), replacing the
// memset + separate += pass.  Pad cols of Hin are 0, bias pads to 0, so pad
// cols of Hout are exactly 0 (K padding for the next WMMA GEMM).
// ---------------------------------------------------------------------------
__global__ void __launch_bounds__(256)
self_bias_init(const float* __restrict__ Hin, const float* __restrict__ dinv,
               const float* __restrict__ bias, int ncols,
               float* __restrict__ Hout)
{
    long long tid = (long long)blockIdx.x * 256 + threadIdx.x;
    int n = (int)(tid >> 6);
    int c = (int)(tid & 63);
    if (n >= N_NODES) return;
    float di = dinv[n];
    float b  = (c < ncols) ? bias[c] : 0.0f;
    Hout[(size_t)n * 64 + c] = Hin[(size_t)n * 64 + c] * di * di + b;
}

// ---------------------------------------------------------------------------
// Edge scatter:  Hout[dst] += Hin[src] * dinv[src]*dinv[dst]
// 16 threads per edge, float4 each (64 floats row stride).
// ---------------------------------------------------------------------------
__global__ void __launch_bounds__(256)
edge_aggregate(const int* __restrict__ ei, const float* __restrict__ dinv,
               const float* __restrict__ Hin, float* __restrict__ Hout)
{
    long long tid = (long long)blockIdx.x * 256 + threadIdx.x;
    long long e   = tid >> 4;
    int grp       = (int)(tid & 15);
    if (e >= N_EDGES) return;
    int s = ei[e];
    int d = ei[N_EDGES + e];
    float coeff = dinv[s] * dinv[d];
    const float4 v = *(const float4*)(Hin + (size_t)s * 64 + grp * 4);
    float* o = Hout + (size_t)d * 64 + grp * 4;
    atomicAdd(o + 0, v.x * coeff);
    atomicAdd(o + 1, v.y * coeff);
    atomicAdd(o + 2, v.z * coeff);
    atomicAdd(o + 3, v.w * coeff);
}

// ---------------------------------------------------------------------------
// global_add_pool: g[batch[n], c] += H[n, c]   (c < 50)
// ---------------------------------------------------------------------------
__global__ void __launch_bounds__(256)
pool_kernel(const float* __restrict__ H, const int* __restrict__ batch,
            float* __restrict__ g)
{
    long long tid = (long long)blockIdx.x * 256 + threadIdx.x;
    int n = (int)(tid >> 6);
    int c = (int)(tid & 63);
    if (n >= N_NODES || c >= 50) return;
    atomicAdd(&g[(size_t)batch[n] * 50 + c], H[(size_t)n * 64 + c]);
}

// ---------------------------------------------------------------------------
// MLP head (all linear, no activations, per the reference): 50 -> 40 -> 20 -> 1
// One thread per graph, weights staged in LDS.
// ---------------------------------------------------------------------------
__global__ void __launch_bounds__(256)
mlp_kernel(const float* __restrict__ g,
           const float* __restrict__ fW1, const float* __restrict__ fb1,
           const float* __restrict__ fW2, const float* __restrict__ fb2,
           const float* __restrict__ fW3, const float* __restrict__ fb3,
           float* __restrict__ out)
{
    __shared__ float W1s[50 * 40];
    __shared__ float W2s[40 * 20];
    __shared__ float W3s[20];
    __shared__ float b1s[40];
    __shared__ float b2s[20];
    const int t = threadIdx.x;
    for (int i = t; i < 50 * 40; i += 256) W1s[i] = fW1[i];
    for (int i = t; i < 40 * 20; i += 256) W2s[i] = fW2[i];
    if (t < 20) W3s[t] = fW3[t];
    if (t < 40) b1s[t] = fb1[t];
    if (t < 20) b2s[t] = fb2[t];
    __syncthreads();

    const int gid = blockIdx.x * 256 + t;
    if (gid >= N_GRAPHS) return;
    const float* row = g + (size_t)gid * 50;

    float t1[40];
    #pragma unroll 4
    for (int o = 0; o < 40; ++o) {
        float acc = b1s[o];
        for (int i = 0; i < 50; ++i) acc += row[i] * W1s[i * 40 + o];
        t1[o] = acc;
    }
    float t2[20];
    #pragma unroll 4
    for (int o = 0; o < 20; ++o) {
        float acc = b2s[o];
        for (int i = 0; i < 40; ++i) acc += t1[i] * W2s[i * 20 + o];
        t2[o] = acc;
    }
    float s = fb3[0];
    for (int i = 0; i < 20; ++i) s += t2[i] * W3s[i];
    out[gid] = s;
}

// ---------------------------------------------------------------------------
// Launch
// ---------------------------------------------------------------------------
extern "C" void kernel_launch(void* const* d_in, const int* in_sizes, int n_in,
                              void* d_out, int out_size, void* d_ws, size_t ws_size,
                              hipStream_t stream) {
    const float* x     = (const float*)d_in[0];
    const int*   ei    = (const int*)  d_in[1];
    const int*   batch = (const int*)  d_in[2];
    const float* W1    = (const float*)d_in[3];
    const float* b1    = (const float*)d_in[4];
    const float* W2    = (const float*)d_in[5];
    const float* b2    = (const float*)d_in[6];
    const float* fW1   = (const float*)d_in[7];
    const float* fb1   = (const float*)d_in[8];
    const float* fW2   = (const float*)d_in[9];
    const float* fb2   = (const float*)d_in[10];
    const float* fW3   = (const float*)d_in[11];
    const float* fb3   = (const float*)d_in[12];
    float* out = (float*)d_out;

    const size_t MB = 1u << 20;
    char* ws = (char*)d_ws;
    float* dinv = (float*)(ws);                 // 0.8 MB
    float* bufA = (float*)(ws + 1 * MB);        // 200000*64*4 = 51.2 MB
    float* bufB = (float*)(ws + 54 * MB);       // 51.2 MB
    float* gbuf = (float*)(ws + 107 * MB);      // 1024*50*4 = 0.2 MB

    const int degBlocks  = (N_EDGES + 255) / 256;                 // 12500
    const int dinvBlocks = (N_NODES + 255) / 256;                 // 782
    const int gemmBlocks = ((N_NODES / 16) + 7) / 8;              // 1563
    const int aggBlocks  = (int)(((long long)N_EDGES * 16 + 255) / 256);   // 200000
    const int nodeBlocks = (int)(((long long)N_NODES * 64 + 255) / 256);   // 50000

    // ---- degree + normalization ----
    hipMemsetAsync(dinv, 0, (size_t)N_NODES * sizeof(float), stream);
    deg_kernel <<<degBlocks, 256, 0, stream>>>(ei, dinv);
    dinv_kernel<<<dinvBlocks, 256, 0, stream>>>(dinv);

    // ---- layer 1: h1 = x @ W1 (WMMA); init self-loop+bias; edge scatter ----
    wmma_gemm_node<<<gemmBlocks, 256, 0, stream>>>(x, N_FEAT, N_FEAT,
                                                   W1, N_FEAT, 60, bufA, N_NODES);
    self_bias_init<<<nodeBlocks, 256, 0, stream>>>(bufA, dinv, b1, 60, bufB);
    edge_aggregate<<<aggBlocks, 256, 0, stream>>>(ei, dinv, bufA, bufB);

    // ---- layer 2: h2 = hA @ W2 (WMMA, K padded 60->64 via zeroed cols) ----
    wmma_gemm_node<<<gemmBlocks, 256, 0, stream>>>(bufB, 64, 64,
                                                   W2, 60, 50, bufA, N_NODES);
    self_bias_init<<<nodeBlocks, 256, 0, stream>>>(bufA, dinv, b2, 50, bufB);
    edge_aggregate<<<aggBlocks, 256, 0, stream>>>(ei, dinv, bufA, bufB);

    // ---- pooling + MLP head ----
    hipMemsetAsync(gbuf, 0, (size_t)N_GRAPHS * 50 * sizeof(float), stream);
    pool_kernel<<<nodeBlocks, 256, 0, stream>>>(bufB, batch, gbuf);
    mlp_kernel <<<(N_GRAPHS + 255) / 256, 256, 0, stream>>>(gbuf, fW1, fb1, fW2, fb2,
                                                            fW3, fb3, out);
}